// PoolingModuleNoProj_33818572489502
// MI455X (gfx1250) — compile-verified
//
#include <hip/hip_runtime.h>

typedef __attribute__((ext_vector_type(2))) float v2f;
typedef __attribute__((ext_vector_type(8))) float v8f;

#define T_DIM 4096
#define N_DIM 16
#define C_DIM 512
#define NC (N_DIM * C_DIM)          // 8192
#define L_CHUNK 16
#define N_CHUNKS (T_DIM / L_CHUNK)  // 256

// ---------------------------------------------------------------------------
// Phase 1: per-chunk column sums.  thread := (chunk, n, c)
// reads 128 MB of x (coalesced: consecutive c -> consecutive addresses),
// writes 8 MB of partials.
// ---------------------------------------------------------------------------
__global__ void chunk_sums_kernel(const float* __restrict__ x,
                                  float* __restrict__ partials) {
    int idx   = blockIdx.x * blockDim.x + threadIdx.x;   // 0 .. CH*N*C-1
    int c     = idx & (C_DIM - 1);
    int n     = (idx >> 9) & (N_DIM - 1);
    int chunk = idx >> 13;

    const float* p = x + (size_t)chunk * L_CHUNK * NC + n * C_DIM + c;
    float s = 0.0f;
#pragma unroll
    for (int j = 0; j < L_CHUNK; ++j) s += p[(size_t)j * NC];
    partials[(size_t)chunk * NC + n * C_DIM + c] = s;
}

// ---------------------------------------------------------------------------
// Phase 2: exclusive scan of the 256 chunk sums per column (hierarchical:
// 8 segments x 32 chunks per block column), producing per-chunk offsets
// that already include cached_avg*cached_len.  Also emits the tail outputs:
// new_cached_len (int32 bits) and new_cached_avg.
// Block := (n, 32-column group); 256 threads = 8 segs x 32 cols.
// ---------------------------------------------------------------------------
__global__ void scan_chunks_kernel(const float* __restrict__ partials,
                                   const int* __restrict__ cached_len,
                                   const float* __restrict__ cached_avg,
                                   float* __restrict__ offsets,
                                   float* __restrict__ out) {
    __shared__ float segsum[8][32];

    const int SEG = N_CHUNKS / 8;       // 32 chunks per segment
    int bid = blockIdx.x;
    int n   = bid >> 4;                 // 16 column-groups per n
    int cg  = bid & 15;
    int cl  = threadIdx.x & 31;
    int seg = threadIdx.x >> 5;
    int c   = cg * 32 + cl;

    const float* p = partials + (size_t)(seg * SEG) * NC + n * C_DIM + c;
    float v[SEG];
    float s = 0.0f;
#pragma unroll
    for (int j = 0; j < SEG; ++j) { v[j] = p[(size_t)j * NC]; s += v[j]; }
    segsum[seg][cl] = s;
    __syncthreads();

    float excl = 0.0f;
#pragma unroll
    for (int si = 0; si < 8; ++si)
        if (si < seg) excl += segsum[si][cl];

    int   len  = cached_len[n];
    float base = cached_avg[n * C_DIM + c] * (float)len;
    float run  = base + excl;

    float* q = offsets + (size_t)(seg * SEG) * NC + n * C_DIM + c;
#pragma unroll
    for (int j = 0; j < SEG; ++j) { q[(size_t)j * NC] = run; run += v[j]; }

    // seg 7 ends with run == base + total sum over all T -> new_cached_avg
    if (seg == 7) {
        float* avg_out = out + (size_t)T_DIM * NC + N_DIM;
        avg_out[n * C_DIM + c] = run * __builtin_amdgcn_rcpf((float)(len + T_DIM));
    }
    if (threadIdx.x == 0 && cg == 0) {
        int* len_out = (int*)(out + (size_t)T_DIM * NC);
        len_out[n] = len + T_DIM;   // int32 bits in the concatenated output
    }
}

// ---------------------------------------------------------------------------
// Phase 3: main pass.  One wave per (chunk, n, 16-column tile).
// Inclusive cumsum over 16 timesteps == L16 (lower-triangular ones) x Xtile,
// done as 4 chained V_WMMA_F32_16X16X4_F32 with the chunk offset preloaded
// into the C accumulator; then per-row rcp(count) scaling on the 8 acc VGPRs
// (v_rcp_f32 + mul: matches the reference's "csum * (1/count)" structure).
// 131072 waves -> saturates HBM; x likely L2-resident from phase 1.
// ---------------------------------------------------------------------------
__global__ void cumsum_wmma_kernel(const float* __restrict__ x,
                                   const float* __restrict__ offsets,
                                   const int* __restrict__ cached_len,
                                   float* __restrict__ out) {
    int lane  = threadIdx.x & 31;
    int w     = blockIdx.x * (blockDim.x >> 5) + (threadIdx.x >> 5);
    int ctile = w & 31;            // C/16 = 32 tiles
    int n     = (w >> 5) & 15;
    int chunk = w >> 9;            // 0..255
    int c0    = ctile * 16;
    int t0    = chunk * 16;

    bool lo  = lane < 16;
    int  col = lane & 15;          // column for B/C/D; row M for A
    int  m   = col;                // A-matrix row index (lanes 0-15 and 16-31 both M=0-15)

    // C accumulator: every row = running offset for this column
    float off = offsets[(size_t)chunk * NC + n * C_DIM + c0 + col];
    v8f acc;
#pragma unroll
    for (int r = 0; r < 8; ++r) acc[r] = off;

    const float* xb = x + (size_t)t0 * NC + n * C_DIM + c0 + col;
    int kbase = lo ? 0 : 2;        // lane-half K split for A and B

#pragma unroll
    for (int s = 0; s < 4; ++s) {
        int k0 = 4 * s + kbase;    // VGPR0 holds K=k0, VGPR1 holds K=k0+1
        int k1 = k0 + 1;
        v2f b;
        b.x = xb[(size_t)k0 * NC]; // B[k][col]: two 64B rows per load
        b.y = xb[(size_t)k1 * NC];
        v2f a;                     // L16[m][k] = (k <= m)
        a.x = (k0 <= m) ? 1.0f : 0.0f;
        a.y = (k1 <= m) ? 1.0f : 0.0f;
        acc = __builtin_amdgcn_wmma_f32_16x16x4_f32(
            /*neg_a=*/false, a, /*neg_b=*/false, b,
            /*c_mod=*/(short)0, acc, /*reuse_a=*/false, /*reuse_b=*/false);
    }

    int len = cached_len[n];
    float* ob = out + (size_t)t0 * NC + n * C_DIM + c0 + col;
    int mbase = lo ? 0 : 8;        // D: VGPR r = rows r (lanes<16) and r+8
    // counts < 2^13 -> exact in fp32; one v_add + v_rcp + v_mul per row
    float cnt0 = (float)(t0 + mbase + 1 + len);
#pragma unroll
    for (int r = 0; r < 8; ++r) {
        float cnt = cnt0 + (float)r;
        ob[(size_t)(r + mbase) * NC] = acc[r] * __builtin_amdgcn_rcpf(cnt);
    }
}

// ---------------------------------------------------------------------------
extern "C" void kernel_launch(void* const* d_in, const int* in_sizes, int n_in,
                              void* d_out, int out_size, void* d_ws, size_t ws_size,
                              hipStream_t stream) {
    const float* x          = (const float*)d_in[0];
    const int*   cached_len = (const int*)d_in[1];
    const float* cached_avg = (const float*)d_in[2];
    float*       out        = (float*)d_out;

    float* partials = (float*)d_ws;                       // 8 MB
    float* offsets  = partials + (size_t)N_CHUNKS * NC;   // 8 MB

    // Phase 1: 256 chunks * 16 n * 512 c = 2,097,152 threads
    chunk_sums_kernel<<<(N_CHUNKS * NC) / 256, 256, 0, stream>>>(x, partials);

    // Phase 2: 16 n * 16 column-groups = 256 blocks of 256
    scan_chunks_kernel<<<N_DIM * (C_DIM / 32), 256, 0, stream>>>(
        partials, cached_len, cached_avg, offsets, out);

    // Phase 3: 256 chunks * 16 n * 32 ctiles = 131072 waves, 8 waves/block
    cumsum_wmma_kernel<<<(N_CHUNKS * N_DIM * 32) / 8, 256, 0, stream>>>(
        x, offsets, cached_len, out);
}